// AngleFreqEnhance_65249143161574
// MI455X (gfx1250) — compile-verified
//
#include <hip/hip_runtime.h>
#include <hip/hip_bf16.h>

// ---------------------------------------------------------------------------
// AngleFreqEnhance for MI455X (gfx1250, wave32).
// Pipeline: gain -> WMMA down-proj (256->16) -> row FFT -> col FFT*gain*col IFFT
//           -> row IFFT -> WMMA up-proj (16->256) + residual.
// Bandwidth-bound (~256MB traffic); WMMA f32 16x16x4 used for both GEMMs.
// ---------------------------------------------------------------------------

typedef __attribute__((ext_vector_type(2))) float v2f;
typedef __attribute__((ext_vector_type(8))) float v8f;

#define HW 16384   // 128*128
#define NPIX 128

__device__ __forceinline__ int bitrev7(int i) {
  return (int)(__brev((unsigned)i) >> 25);
}

// 7-stage radix-2 DIT FFT over 128 complex points in LDS; 64 threads, 1
// butterfly each per stage. Input must already be in bit-reversed order.
// sign = -1 forward, +1 inverse (unnormalized).
__device__ __forceinline__ void fft_stages(float2* s, int t, float sign) {
  const float PI = 3.14159265358979323846f;
#pragma unroll
  for (int st = 0; st < 7; ++st) {
    __syncthreads();
    const int half = 1 << st;
    const int pos = t & (half - 1);
    const int i0 = ((t >> st) << (st + 1)) + pos;
    const int i1 = i0 + half;
    const float ang = sign * PI * (float)pos / (float)half;
    float sw, cw;
    __sincosf(ang, &sw, &cw);
    const float2 a = s[i0];
    const float2 b = s[i1];
    const float br = b.x * cw - b.y * sw;
    const float bi = b.x * sw + b.y * cw;
    s[i0] = make_float2(a.x + br, a.y + bi);
    s[i1] = make_float2(a.x - br, a.y - bi);
  }
}

// gain[c][h][w] in UNSHIFTED spectrum coords (fftshift folded in).
__global__ void afe_gain(const float* __restrict__ bw, float* __restrict__ gain) {
  const int p = blockIdx.x * blockDim.x + threadIdx.x;  // 0..16383
  if (p >= HW) return;
  const int h = p >> 7, w = p & 127;
  const float fy = (h < 64) ? (float)h : (float)(h - 128);
  const float fx = (w < 64) ? (float)w : (float)(w - 128);
  const float r = sqrtf(fy * fy + fx * fx);
  int ridx = (int)floorf(r * 0.125f);          // RADIUS_WIDTH = 8
  if (ridx > 8) ridx = 8;                      // n_radii-1 = 8
  const float PI = 3.14159265358979323846f;
  float theta = fmodf(atan2f(fy, fx) + PI, PI);
  const float delta = PI / 8.0f;               // N_ANGLES = 8
  const float hwid = 0.75f * delta;            // OVERLAP_RATIO*delta/2
  float aw[8];
  float sum = 0.f;
#pragma unroll
  for (int a = 0; a < 8; ++a) {
    const float c = ((float)a + 0.5f) * delta;
    const float d = fabsf(theta - c);
    const float wv = fmaxf(1.f - d / hwid, 0.f);
    aw[a] = wv;
    sum += wv;
  }
  const float inv = 1.f / (sum + 1e-8f);
#pragma unroll
  for (int c = 0; c < 16; ++c) {
    float g = 0.f;
#pragma unroll
    for (int a = 0; a < 8; ++a) g += aw[a] * bw[c * 72 + a * 9 + ridx];
    gain[c * HW + p] = g * inv;
  }
}

// Down-projection: xproj[b,m,hw] = sum_c w_in[m,c] * x[b,c,hw].
// One wave = one 16(m) x 16(pixel) tile, 64 chained v_wmma_f32_16x16x4_f32.
__global__ void __launch_bounds__(256) afe_proj_down(const float* __restrict__ x,
                                                     const float* __restrict__ w_in,
                                                     float* __restrict__ xproj) {
  __shared__ float sw[16 * 256];
  for (int i = threadIdx.x; i < 16 * 256; i += 256) sw[i] = w_in[i];
  __syncthreads();
  const int lane = threadIdx.x & 31;
  const int wave = threadIdx.x >> 5;
  const int group = blockIdx.x * 8 + wave;       // 16-pixel group, 8192 total
  const int p0 = group * 16;
  const int b = p0 / HW;
  const int rem = p0 - b * HW;
  const int n = lane & 15;
  const int kh = lane >> 4;                      // K half-select per ISA layout
  const float* wrow = sw + n * 256 + 2 * kh;     // A: M=lane&15, K=k0+2*kh(+1)
  const float* xb = x + (size_t)b * 256 * HW + rem + n;

  v8f acc = {};
  for (int k0 = 0; k0 < 256; k0 += 4) {
    v2f a, bb;
    a.x = wrow[k0];
    a.y = wrow[k0 + 1];
    const float* xp = xb + (size_t)(k0 + 2 * kh) * HW;  // B: K=k0+2*kh(+1), N=n
    bb.x = xp[0];
    bb.y = xp[HW];
    acc = __builtin_amdgcn_wmma_f32_16x16x4_f32(false, a, false, bb, (short)0,
                                                acc, false, false);
  }
  float* outp = xproj + (size_t)b * 16 * HW + rem + n;
#pragma unroll
  for (int v = 0; v < 8; ++v) {                  // D: M = v + 8*kh, N = lane&15
    outp[(size_t)(v + 8 * kh) * HW] = acc[v];
  }
}

// Row FFT: real -> complex (unnormalized forward).
__global__ void afe_fft_rows(const float* __restrict__ in, float2* __restrict__ out) {
  __shared__ float2 s[NPIX];
  const int t = threadIdx.x;                     // 0..63
  const size_t base = (size_t)blockIdx.x * NPIX;
  s[bitrev7(t)] = make_float2(in[base + t], 0.f);
  s[bitrev7(t + 64)] = make_float2(in[base + t + 64], 0.f);
  fft_stages(s, t, -1.f);
  __syncthreads();
  out[base + t] = s[t];
  out[base + t + 64] = s[t + 64];
}

// Column FFT -> spectral gain -> column IFFT, in place per column.
__global__ void afe_fft_cols_gain(float2* __restrict__ data,
                                  const float* __restrict__ gain) {
  __shared__ float2 s[NPIX];
  __shared__ float2 s2[NPIX];
  const int t = threadIdx.x;
  const int img = blockIdx.x >> 7;               // b*16 + m
  const int w = blockIdx.x & 127;
  const int m = img & 15;
  float2* col = data + (size_t)img * HW + w;
  s[bitrev7(t)] = col[(size_t)t * NPIX];
  s[bitrev7(t + 64)] = col[(size_t)(t + 64) * NPIX];
  fft_stages(s, t, -1.f);
  __syncthreads();
#pragma unroll
  for (int e = 0; e < 2; ++e) {
    const int h = t + 64 * e;
    const float2 F = s[h];
    const float mag = sqrtf(F.x * F.x + F.y * F.y);
    const float g = gain[m * HW + h * NPIX + w];
    // ortho-domain EPS rescaled: |F_ortho| = |F|/128 -> EPS*128
    const float sc = g * (mag + 1.28e-6f) / fmaxf(mag, 1e-30f);
    s2[bitrev7(h)] = make_float2(F.x * sc, F.y * sc);
  }
  fft_stages(s2, t, +1.f);
  __syncthreads();
  col[(size_t)t * NPIX] = s2[t];
  col[(size_t)(t + 64) * NPIX] = s2[t + 64];
}

// Row IFFT -> real part, with the full 1/(128*128) ortho normalization.
__global__ void afe_ifft_rows(const float2* __restrict__ in, float* __restrict__ out) {
  __shared__ float2 s[NPIX];
  const int t = threadIdx.x;
  const size_t base = (size_t)blockIdx.x * NPIX;
  s[bitrev7(t)] = in[base + t];
  s[bitrev7(t + 64)] = in[base + t + 64];
  fft_stages(s, t, +1.f);
  __syncthreads();
  const float inv = 1.0f / 16384.0f;
  out[base + t] = s[t].x * inv;
  out[base + t + 64] = s[t + 64].x * inv;
}

// Up-projection + residual: out[b,c,hw] = x[b,c,hw] + sum_m w_out[c,m]*xenh[b,m,hw].
// One wave = 16(c) x 16(pixel) tile, 4 chained WMMAs over K=16.
__global__ void __launch_bounds__(256) afe_proj_up(const float* __restrict__ x,
                                                   const float* __restrict__ w_out,
                                                   const float* __restrict__ xenh,
                                                   float* __restrict__ out) {
  __shared__ float sw[256 * 16];
  for (int i = threadIdx.x; i < 256 * 16; i += 256) sw[i] = w_out[i];
  __syncthreads();
  const int lane = threadIdx.x & 31;
  const int wave = threadIdx.x >> 5;
  const int wid = blockIdx.x * 8 + wave;         // 131072 waves
  const int ct = wid & 15;                       // channel tile (16 per group)
  const int pg = wid >> 4;                       // 16-pixel group
  const int p0 = pg * 16;
  const int b = p0 / HW;
  const int rem = p0 - b * HW;
  const int n = lane & 15;
  const int kh = lane >> 4;
  const int c0 = ct * 16;
  const float* wrow = sw + (c0 + n) * 16 + 2 * kh;
  const float* eb = xenh + (size_t)b * 16 * HW + rem + n;
  const float* xr = x + (size_t)b * 256 * HW + (size_t)(c0 + n) * HW + rem;
  __builtin_prefetch(xr, 0, 0);                  // warm residual line (global_prefetch)

  v8f acc = {};
#pragma unroll
  for (int k0 = 0; k0 < 16; k0 += 4) {
    v2f a, bb;
    a.x = wrow[k0];
    a.y = wrow[k0 + 1];
    const float* ep = eb + (size_t)(k0 + 2 * kh) * HW;
    bb.x = ep[0];
    bb.y = ep[HW];
    acc = __builtin_amdgcn_wmma_f32_16x16x4_f32(false, a, false, bb, (short)0,
                                                acc, false, false);
  }
  const size_t obase = (size_t)b * 256 * HW + (size_t)c0 * HW + rem + n;
#pragma unroll
  for (int v = 0; v < 8; ++v) {
    const size_t idx = obase + (size_t)(v + 8 * kh) * HW;
    out[idx] = x[idx] + acc[v];
  }
}

extern "C" void kernel_launch(void* const* d_in, const int* in_sizes, int n_in,
                              void* d_out, int out_size, void* d_ws, size_t ws_size,
                              hipStream_t stream) {
  const float* x = (const float*)d_in[0];        // (8,256,128,128)
  const float* w_in = (const float*)d_in[1];     // (16,256)
  const float* w_out = (const float*)d_in[2];    // (256,16)
  const float* bw = (const float*)d_in[3];       // (16,8,9)
  float* out = (float*)d_out;

  float* ws = (float*)d_ws;
  float* gain = ws;                              // 262144 f32  (1 MB)
  float* realbuf = ws + 262144;                  // 2097152 f32 (8 MB): x_proj / x_enh
  float2* cbuf = (float2*)(ws + 262144 + 2097152);  // 2097152 cpx (16 MB)

  afe_gain<<<64, 256, 0, stream>>>(bw, gain);
  afe_proj_down<<<1024, 256, 0, stream>>>(x, w_in, realbuf);
  afe_fft_rows<<<16384, 64, 0, stream>>>(realbuf, cbuf);
  afe_fft_cols_gain<<<16384, 64, 0, stream>>>(cbuf, gain);
  afe_ifft_rows<<<16384, 64, 0, stream>>>(cbuf, realbuf);
  afe_proj_up<<<16384, 256, 0, stream>>>(x, w_out, realbuf, out);
}